// Net_19945828122986
// MI455X (gfx1250) — compile-verified
//
#include <hip/hip_runtime.h>
#include <hip/hip_bf16.h>

#define USE_TDM 1

typedef float v2f __attribute__((ext_vector_type(2)));
typedef float v4f __attribute__((ext_vector_type(4)));
typedef float v8f __attribute__((ext_vector_type(8)));
typedef unsigned int u32x4 __attribute__((ext_vector_type(4)));
typedef int          i32x8 __attribute__((ext_vector_type(8)));
typedef int          i32x4 __attribute__((ext_vector_type(4)));

#define DIM            32
#define KTOT           1024                 // 32*32 flattened contraction length
#define ROWS_PER_WAVE  16                   // WMMA M
#define WAVES_PER_BLK  8
#define THREADS        (WAVES_PER_BLK * 32)
#define ROWS_PER_BLK   (ROWS_PER_WAVE * WAVES_PER_BLK)   // 128
#define STAGE_F        64                   // floats per row staged per LDS stage
#define NSTAGE         (KTOT / STAGE_F)     // 16
#define CHUNKS_PER_ST  (STAGE_F / 4)        // 16 WMMAs (K=4) per stage
#define LDS_STRIDE     68                   // padded row stride (floats): 68%64==4 -> conflict-free frags
#define TILE_F         (ROWS_PER_WAVE * LDS_STRIDE)      // 1088 floats per buffer
#define NCHUNK         (KTOT / 4)           // 256 K-chunks total

// ---------------- circuit-build helpers (tiny one-time kernel) ----------------

__device__ inline void apply_ry(float* re, float* im, int w, float th) {
    float c = cosf(0.5f * th), s = sinf(0.5f * th);
    int p = 4 - w, mask = 1 << p;
    for (int i = 0; i < 32; ++i) {
        if (i & mask) continue;
        int j = i | mask;
        float ar = re[i], ai = im[i], br = re[j], bi = im[j];
        re[i] = c * ar - s * br;  im[i] = c * ai - s * bi;
        re[j] = s * ar + c * br;  im[j] = s * ai + c * bi;
    }
}

__device__ inline void apply_rx(float* re, float* im, int w, float th) {
    float c = cosf(0.5f * th), s = sinf(0.5f * th);
    int p = 4 - w, mask = 1 << p;
    for (int i = 0; i < 32; ++i) {
        if (i & mask) continue;
        int j = i | mask;
        float ar = re[i], ai = im[i], br = re[j], bi = im[j];
        // RX = [[c, -i s], [-i s, c]]
        re[i] = c * ar + s * bi;  im[i] = c * ai - s * br;
        re[j] = c * br + s * ai;  im[j] = c * bi - s * ar;
    }
}

__device__ inline void apply_cnot(float* re, float* im, int cq, int tq) {
    int pc = 4 - cq, pt = 4 - tq, mt = 1 << pt;
    for (int i = 0; i < 32; ++i) {
        if (!((i >> pc) & 1)) continue;   // control bit set
        if (i & mt) continue;             // visit each pair once
        int j = i | mt;
        float t;
        t = re[i]; re[i] = re[j]; re[j] = t;
        t = im[i]; im[i] = im[j]; im[j] = t;
    }
}

__device__ inline float wval(int n, int f,
                             const float* u0r, const float* u0i,
                             const float* u1r, const float* u1i,
                             const float* pr,  const float* pi) {
    int j = f >> 5, k = f & 31;
    float ur = (n < 2) ? u0r[j] : u1r[j];
    float ui = (n < 2) ? u0i[j] : u1i[j];
    float prodr = ur * pr[k] - ui * pi[k];
    float prodi = ur * pi[k] + ui * pr[k];
    return (n & 1) ? prodi : prodr;
}

// Builds psi = U1[:,0], rows 0/1 of U2, then pre-swizzled WMMA B-fragments.
__global__ void build_weights(const float* __restrict__ p1,
                              const float* __restrict__ p2,
                              float2* __restrict__ wfrag) {
    __shared__ float su0r[32], su0i[32], su1r[32], su1i[32], spr[32], spi[32];
    int tid = threadIdx.x;
    float re[32], im[32];
    for (int i = 0; i < 32; ++i) { re[i] = 0.f; im[i] = 0.f; }

    if (tid < 32) {
        re[tid] = 1.f;                        // circuit 2 column = basis state tid
        for (int b = 0; b < 4; ++b) {
            for (int w = 0; w < 4; ++w) apply_ry(re, im, w, p2[(b * 2 + 0) * 4 + w]);
            for (int w = 0; w < 4; ++w) apply_rx(re, im, w, p2[(b * 2 + 1) * 4 + w]);
            for (int t = 0; t < 4; ++t) if (t != b) apply_cnot(re, im, b, t);
        }
        su0r[tid] = re[0]; su0i[tid] = im[0];   // U2[0, tid]
        su1r[tid] = re[1]; su1i[tid] = im[1];   // U2[1, tid]
    } else if (tid == 32) {
        re[0] = 1.f;                          // circuit 1 on |00000>
        for (int b = 0; b < 5; ++b) {
            for (int w = 0; w < 5; ++w) apply_ry(re, im, w, p1[(b * 2 + 0) * 5 + w]);
            for (int w = 0; w < 5; ++w) apply_rx(re, im, w, p1[(b * 2 + 1) * 5 + w]);
            for (int t = 0; t < 5; ++t) if (t != b) apply_cnot(re, im, b, t);
        }
        for (int i = 0; i < 32; ++i) { spr[i] = re[i]; spi[i] = im[i]; }
    }
    __syncthreads();

    for (int e = tid; e < NCHUNK * 32; e += 64) {
        int c = e >> 5, l = e & 31;
        int n  = l & 15;
        int k0 = (l >> 4) * 2;
        float2 v = make_float2(0.f, 0.f);
        if (n < 4) {
            int f0 = 4 * c + k0;
            v.x = wval(n, f0,     su0r, su0i, su1r, su1i, spr, spi);
            v.y = wval(n, f0 + 1, su0r, su0i, su1r, su1i, spr, spi);
        }
        wfrag[e] = v;
    }
}

// ---------------- TDM issue: one DMA per wave per stage ----------------
// D# describes a 2D tile: 16 rows x 64 fp32, row stride 1024 elements in memory,
// with LDS padding (64 DWORDs then +4 DWORDs) -> LDS row stride 68 floats,
// which makes the A-fragment ds_load_b64 pattern bank-conflict-free.

#if USE_TDM
__device__ inline void tdm_issue(unsigned lds_off, unsigned long long gaddr) {
    u32x4 g0;
    g0.x = 1u;                                           // count=1 user descriptor
    g0.y = lds_off;                                      // LDS byte address
    g0.z = (unsigned)(gaddr & 0xFFFFFFFFull);            // global_addr[31:0]
    g0.w = (unsigned)((gaddr >> 32) & 0x01FFFFFFull)     // global_addr[56:32]
         | (2u << 30);                                   // type = 2 ("image")

    i32x8 g1;
    g1[0] = (2 << 16)        // data_size = 4 bytes
          | (1 << 20)        // pad_enable
          | (5 << 22)        // pad_interval: 64 DWORDs
          | (3 << 25);       // pad_amount: 4 DWORDs
    g1[1] = (int)(1024u << 16);                 // tensor_dim0[15:0]=1024 (atomic bar addr=0)
    g1[2] = 0;                                  // tensor_dim0[31:16]=0 | tensor_dim1[15:0]=0
    g1[3] = (int)(((1u << 24) >> 16))           // tensor_dim1[31:16] (dim1 = 2^24, no OOB)
          | (64 << 16);                         // tile_dim0 = 64 elements
    g1[4] = 16;                                 // tile_dim1 = 16 rows, tile_dim2 = 0
    g1[5] = 1024;                               // tensor_dim0_stride[31:0] = 1024 elements
    g1[6] = 0;                                  // stride0 hi | stride1 lo
    g1[7] = 0;                                  // stride1 hi

    i32x4 z4 = {0, 0, 0, 0};                    // groups 2/3 unused (2D tile)
    i32x8 z8 = {0, 0, 0, 0, 0, 0, 0, 0};        // extra group (clang-23 6-arg form)
    __builtin_amdgcn_tensor_load_to_lds(g0, g1, z4, z4, z8, 0);
}
#endif

// ---------------- hot kernel: (B x 1024) · (1024 x 4) via WMMA f32 ----------------

__global__ __launch_bounds__(THREADS) void oracle_contract(
    const float* __restrict__ O,          // [B][1024] row-major fp32
    const float2* __restrict__ wfrag,     // [256][32] pre-swizzled B fragments (L2-resident)
    float2* __restrict__ out) {           // [B] -> (x, 1-x)
    __shared__ alignas(16) float tile[WAVES_PER_BLK][2][TILE_F];

    const int lane = threadIdx.x & 31;
    const unsigned wv = (unsigned)__builtin_amdgcn_readfirstlane((int)(threadIdx.x >> 5));
    const unsigned long long b0 = (unsigned long long)blockIdx.x * ROWS_PER_BLK
                                + (unsigned long long)wv * ROWS_PER_WAVE;

    v8f acc = {0.f, 0.f, 0.f, 0.f, 0.f, 0.f, 0.f, 0.f};
    const int m    = lane & 15;           // A-matrix row this lane serves
    const int koff = (lane >> 4) * 2;     // K 0/1 (lanes 0-15) vs K 2/3 (lanes 16-31)

#if USE_TDM
    const unsigned lds0 = (unsigned)(size_t)&tile[wv][0][0];
    const unsigned lds1 = (unsigned)(size_t)&tile[wv][1][0];
    const unsigned long long gbase = (unsigned long long)(size_t)O + b0 * (KTOT * 4ull);

    tdm_issue(lds0, gbase);                                  // prologue: stage 0
    for (int s = 0; s < NSTAGE; ++s) {
        asm volatile("" ::: "memory");
        if (s + 1 < NSTAGE) {                                // prefetch stage s+1 via TDM
            tdm_issue(((s + 1) & 1) ? lds1 : lds0,
                      gbase + (unsigned long long)(s + 1) * (STAGE_F * 4ull));
            __builtin_amdgcn_s_wait_tensorcnt(1);            // stage s DMA complete
        } else {
            __builtin_amdgcn_s_wait_tensorcnt(0);
        }
        asm volatile("" ::: "memory");

        const float* t = &tile[wv][s & 1][0];
        #pragma unroll
        for (int cl = 0; cl < CHUNKS_PER_ST; ++cl) {
            int c = s * CHUNKS_PER_ST + cl;
            v2f a;                                   // A 16x4 fp32 fragment (ds_load_b64)
            a.x = t[m * LDS_STRIDE + cl * 4 + koff];
            a.y = t[m * LDS_STRIDE + cl * 4 + koff + 1];
            float2 bw = wfrag[c * 32 + lane];        // B 4x16 fragment
            v2f b; b.x = bw.x; b.y = bw.y;
            acc = __builtin_amdgcn_wmma_f32_16x16x4_f32(
                false, a, false, b, (short)0, acc, false, false);
        }
    }
#else
    const v4f* Ov = (const v4f*)O;
    float* t = &tile[wv][0][0];
    for (int s = 0; s < NSTAGE; ++s) {
        #pragma unroll
        for (int i = 0; i < 8; ++i) {
            int  r  = i * 2 + (lane >> 4);
            unsigned long long gi = (b0 + r) * (KTOT / 4) + (unsigned long long)s * (STAGE_F / 4) + (lane & 15);
            v4f v = __builtin_nontemporal_load(&Ov[gi]);
            *(v4f*)&t[r * LDS_STRIDE + (lane & 15) * 4] = v;
        }
        __syncthreads();
        #pragma unroll
        for (int cl = 0; cl < CHUNKS_PER_ST; ++cl) {
            int c = s * CHUNKS_PER_ST + cl;
            v2f a;
            a.x = t[m * LDS_STRIDE + cl * 4 + koff];
            a.y = t[m * LDS_STRIDE + cl * 4 + koff + 1];
            float2 bw = wfrag[c * 32 + lane];
            v2f b; b.x = bw.x; b.y = bw.y;
            acc = __builtin_amdgcn_wmma_f32_16x16x4_f32(
                false, a, false, b, (short)0, acc, false, false);
        }
        __syncthreads();
    }
#endif

    // D layout: VGPR r holds row m=r (lanes 0-15) / m=r+8 (lanes 16-31), N=lane&15.
    // x = sum over N=0..3 of D^2 (columns 4..15 accumulated zeros).
    #pragma unroll
    for (int r = 0; r < 8; ++r) {
        float v  = acc[r];
        float sq = v * v;
        sq += __shfl_xor(sq, 1, 32);
        sq += __shfl_xor(sq, 2, 32);
        if ((lane & 15) == 0) {
            int row = r + ((lane >> 4) << 3);
            float2 o; o.x = sq; o.y = 1.0f - sq;
            out[b0 + row] = o;
        }
    }
}

// ---------------- launch ----------------

extern "C" void kernel_launch(void* const* d_in, const int* in_sizes, int n_in,
                              void* d_out, int out_size, void* d_ws, size_t ws_size,
                              hipStream_t stream) {
    const float* oracles = (const float*)d_in[0];   // (B, 32, 32) fp32
    const float* params1 = (const float*)d_in[1];   // (5, 2, 5)  fp32
    const float* params2 = (const float*)d_in[2];   // (4, 2, 4)  fp32

    float2* wfrag = (float2*)d_ws;                  // 256*32 float2 = 64 KB scratch

    int B = in_sizes[0] / (DIM * DIM);              // 65536

    build_weights<<<1, 64, 0, stream>>>(params1, params2, wfrag);

    int blocks = B / ROWS_PER_BLK;                  // 512
    oracle_contract<<<blocks, THREADS, 0, stream>>>(oracles, wfrag, (float2*)d_out);
}